// CodeGraphGNN_53377853555467
// MI455X (gfx1250) — compile-verified
//
#include <hip/hip_runtime.h>
#include <math.h>

typedef float v2f __attribute__((ext_vector_type(2)));
typedef float v8f __attribute__((ext_vector_type(8)));

// ---------------------------------------------------------------------------
// WMMA GEMM: HS[row, 0..N) = (Hin[row, 0..K) @ W[K x N]) * dis[row]
// One wave per 16-row tile. f32 16x16x4 WMMA, K stepped by 4.
// A-frag (16x4): lane m=lane&15 is row M; lanes<16 carry K+0/K+1, lanes>=16 K+2/K+3.
// B-frag (4x16): lane m is col N; same K split across half-waves.
// C/D (16x16):   VGPR v holds rows v (lanes 0-15) and v+8 (lanes 16-31).
// K-tail handling is branchless: clamped (always in-bounds) loads * {0,1} mask,
// so no EXEC save/restore blocks appear and loads can be clustered.
// ---------------------------------------------------------------------------
template <int K, int N, int LDA>
__global__ void gcn_gemm_wmma(const float* __restrict__ Hin,
                              const float* __restrict__ W,
                              const float* __restrict__ dis,
                              float* __restrict__ HS,
                              int Mtiles)
{
    constexpr int NT = N / 16;
    const int wave = threadIdx.x >> 5;
    const int lane = threadIdx.x & 31;
    const int tile = blockIdx.x * (blockDim.x >> 5) + wave;
    if (tile >= Mtiles) return;              // wave-uniform: EXEC stays all-1s for WMMA

    const int row0 = tile << 4;
    const int m    = lane & 15;
    const bool hi  = lane >= 16;

    v8f acc[NT];
    #pragma unroll
    for (int nt = 0; nt < NT; ++nt) acc[nt] = (v8f){0,0,0,0,0,0,0,0};

    const float* arow = Hin + (size_t)(row0 + m) * LDA;

    #pragma unroll
    for (int k0 = 0; k0 < K; k0 += 4) {
        const int ka  = k0 + (hi ? 2 : 0);
        const int k1  = ka + 1;
        // Branchless tail: clamp address into bounds, zero via mask.
        const int   kac = (ka < K) ? ka : (K - 1);
        const int   k1c = (k1 < K) ? k1 : (K - 1);
        const float m0  = (ka < K) ? 1.0f : 0.0f;
        const float m1  = (k1 < K) ? 1.0f : 0.0f;
        v2f A;
        A.x = arow[kac] * m0;
        A.y = arow[k1c] * m1;
        #pragma unroll
        for (int nt = 0; nt < NT; ++nt) {
            const int col = nt * 16 + m;
            v2f B;
            B.x = W[(size_t)kac * N + col] * m0;
            B.y = W[(size_t)k1c * N + col] * m1;
            acc[nt] = __builtin_amdgcn_wmma_f32_16x16x4_f32(
                false, A, false, B, (short)0, acc[nt], false, false);
        }
    }

    // Epilogue: scale each output row by dis[row] and store HS.
    const int rbase = row0 + (hi ? 8 : 0);
    float d[8];
    #pragma unroll
    for (int v = 0; v < 8; ++v) d[v] = dis[rbase + v];
    #pragma unroll
    for (int nt = 0; nt < NT; ++nt) {
        #pragma unroll
        for (int v = 0; v < 8; ++v) {
            HS[(size_t)(rbase + v) * N + nt * 16 + m] = acc[nt][v] * d[v];
        }
    }
}

// ---------------------------------------------------------------------------
// Degree / normalization helpers
// ---------------------------------------------------------------------------
__global__ void fill_val(float* __restrict__ p, float v, int n)
{
    int i = blockIdx.x * blockDim.x + threadIdx.x;
    if (i < n) p[i] = v;
}

__global__ void deg_count(const int* __restrict__ dst, float* __restrict__ deg, int E)
{
    int e = blockIdx.x * blockDim.x + threadIdx.x;
    if (e < E) atomicAdd(&deg[dst[e]], 1.0f);
}

__global__ void rsqrt_inplace(float* __restrict__ p, int n)
{
    int i = blockIdx.x * blockDim.x + threadIdx.x;
    if (i < n) p[i] = rsqrtf(p[i]);
}

// ---------------------------------------------------------------------------
// Edge scatter: ACC[dst] += HS[src]   (wave32 per edge; contiguous row move)
// ---------------------------------------------------------------------------
__global__ void scatter_add64(const int* __restrict__ src, const int* __restrict__ dst,
                              const float* __restrict__ hs, float* __restrict__ acc, int E)
{
    const int wave = threadIdx.x >> 5, lane = threadIdx.x & 31;
    const int e = blockIdx.x * (blockDim.x >> 5) + wave;
    if (e >= E) return;
    const int s = src[e], d = dst[e];
    const float2 v = ((const float2*)(hs + (size_t)s * 64))[lane];
    float* a = acc + (size_t)d * 64 + lane * 2;
    atomicAdd(a,     v.x);
    atomicAdd(a + 1, v.y);
}

__global__ void scatter_add32(const int* __restrict__ src, const int* __restrict__ dst,
                              const float* __restrict__ hs, float* __restrict__ acc, int E)
{
    const int wave = threadIdx.x >> 5, lane = threadIdx.x & 31;
    const int e = blockIdx.x * (blockDim.x >> 5) + wave;
    if (e >= E) return;
    const int s = src[e], d = dst[e];
    const float v = hs[(size_t)s * 32 + lane];
    atomicAdd(acc + (size_t)d * 32 + lane, v);
}

// ---------------------------------------------------------------------------
// Combine: H[i,f] = relu(dis[i]*(ACC[i,f] + HS[i,f]) + b[f])   (in place in ACC)
// ---------------------------------------------------------------------------
__global__ void combine_relu(float* __restrict__ acch, const float* __restrict__ hs,
                             const float* __restrict__ dis, const float* __restrict__ bias,
                             int total, int logF)
{
    int idx = blockIdx.x * blockDim.x + threadIdx.x;
    if (idx >= total) return;
    const int f = idx & ((1 << logF) - 1);
    const int i = idx >> logF;
    float v = dis[i] * (acch[idx] + hs[idx]) + bias[f];
    acch[idx] = v > 0.0f ? v : 0.0f;
}

// ---------------------------------------------------------------------------
// Classifier: logits = H3[i,0:32] @ Wc[32x3] + bc ; softmax.  Wave32 per node.
// ---------------------------------------------------------------------------
__global__ void classify_softmax(const float* __restrict__ h, const float* __restrict__ Wc,
                                 const float* __restrict__ bc, float* __restrict__ out, int N)
{
    const int wave = threadIdx.x >> 5, lane = threadIdx.x & 31;
    const int node = blockIdx.x * (blockDim.x >> 5) + wave;
    if (node >= N) return;
    const float hv = h[(size_t)node * 32 + lane];
    float p0 = hv * Wc[lane * 3 + 0];
    float p1 = hv * Wc[lane * 3 + 1];
    float p2 = hv * Wc[lane * 3 + 2];
    #pragma unroll
    for (int off = 16; off > 0; off >>= 1) {
        p0 += __shfl_xor(p0, off, 32);
        p1 += __shfl_xor(p1, off, 32);
        p2 += __shfl_xor(p2, off, 32);
    }
    if (lane == 0) {
        p0 += bc[0]; p1 += bc[1]; p2 += bc[2];
        const float mx = fmaxf(p0, fmaxf(p1, p2));
        const float e0 = __expf(p0 - mx), e1 = __expf(p1 - mx), e2 = __expf(p2 - mx);
        const float inv = 1.0f / (e0 + e1 + e2);
        out[(size_t)node * 3 + 0] = e0 * inv;
        out[(size_t)node * 3 + 1] = e1 * inv;
        out[(size_t)node * 3 + 2] = e2 * inv;
    }
}

// ---------------------------------------------------------------------------
extern "C" void kernel_launch(void* const* d_in, const int* in_sizes, int n_in,
                              void* d_out, int out_size, void* d_ws, size_t ws_size,
                              hipStream_t stream)
{
    const float* x  = (const float*)d_in[0];
    const int*   ei = (const int*)  d_in[1];
    const float* W1 = (const float*)d_in[2]; const float* b1 = (const float*)d_in[3];
    const float* W2 = (const float*)d_in[4]; const float* b2 = (const float*)d_in[5];
    const float* W3 = (const float*)d_in[6]; const float* b3 = (const float*)d_in[7];
    const float* Wc = (const float*)d_in[8]; const float* bc = (const float*)d_in[9];
    float* out = (float*)d_out;

    const int N = in_sizes[0] / 10;   // 100000
    const int E = in_sizes[1] / 2;    // 3200000
    const int* src = ei;
    const int* dst = ei + E;

    // Workspace layout (floats): dis[N], then three N*64 rotating buffers.
    float* ws   = (float*)d_ws;
    float* dis  = ws;
    float* buf0 = ws + 102400;                 // HS scratch
    float* buf1 = buf0 + (size_t)N * 64;       // ACC / H state
    float* buf2 = buf1 + (size_t)N * 64;       // ACC / H state

    const int BLK = 256;
    const int Mtiles     = (N + 15) / 16;
    const int gemmBlocks = (Mtiles + 7) / 8;
    const int edgeBlocks = (E + 7) / 8;        // wave32 per edge, 8 waves/block
    const int nodeBlocks = (N + 7) / 8;

    // --- degree + dis = rsqrt(indeg + 1) ---
    fill_val<<<(N + BLK - 1) / BLK, BLK, 0, stream>>>(dis, 1.0f, N);
    deg_count<<<(E + BLK - 1) / BLK, BLK, 0, stream>>>(dst, dis, E);
    rsqrt_inplace<<<(N + BLK - 1) / BLK, BLK, 0, stream>>>(dis, N);

    // --- layer 1: x[100000x10] @ W1[10x64] ---
    gcn_gemm_wmma<10, 64, 10><<<gemmBlocks, BLK, 0, stream>>>(x, W1, dis, buf0, Mtiles);
    fill_val<<<((N * 64) + BLK - 1) / BLK, BLK, 0, stream>>>(buf1, 0.0f, N * 64);
    scatter_add64<<<edgeBlocks, BLK, 0, stream>>>(src, dst, buf0, buf1, E);
    combine_relu<<<((N * 64) + BLK - 1) / BLK, BLK, 0, stream>>>(buf1, buf0, dis, b1, N * 64, 6);

    // --- layer 2: H1 @ W2[64x64] ---
    gcn_gemm_wmma<64, 64, 64><<<gemmBlocks, BLK, 0, stream>>>(buf1, W2, dis, buf0, Mtiles);
    fill_val<<<((N * 64) + BLK - 1) / BLK, BLK, 0, stream>>>(buf2, 0.0f, N * 64);
    scatter_add64<<<edgeBlocks, BLK, 0, stream>>>(src, dst, buf0, buf2, E);
    combine_relu<<<((N * 64) + BLK - 1) / BLK, BLK, 0, stream>>>(buf2, buf0, dis, b2, N * 64, 6);

    // --- layer 3: H2 @ W3[64x32] ---
    gcn_gemm_wmma<64, 32, 64><<<gemmBlocks, BLK, 0, stream>>>(buf2, W3, dis, buf0, Mtiles);
    fill_val<<<((N * 32) + BLK - 1) / BLK, BLK, 0, stream>>>(buf1, 0.0f, N * 32);
    scatter_add32<<<edgeBlocks, BLK, 0, stream>>>(src, dst, buf0, buf1, E);
    combine_relu<<<((N * 32) + BLK - 1) / BLK, BLK, 0, stream>>>(buf1, buf0, dis, b3, N * 32, 5);

    // --- classifier + softmax ---
    classify_softmax<<<nodeBlocks, BLK, 0, stream>>>(buf1, Wc, bc, out, N);
}